// LQR_57166014710456
// MI455X (gfx1250) — compile-verified
//
#include <hip/hip_runtime.h>
#include <hip/hip_bf16.h>

#define NBATCH 32
#define TT 256
#define NSd 64
#define NCd 32
#define NSCd 96

typedef float v2f __attribute__((ext_vector_type(2)));
typedef float v8f __attribute__((ext_vector_type(8)));
typedef float v2fa __attribute__((ext_vector_type(2), aligned(8)));
typedef unsigned int u32x4 __attribute__((ext_vector_type(4)));
typedef int i32x4 __attribute__((ext_vector_type(4)));
typedef int i32x8 __attribute__((ext_vector_type(8)));

// ---- TDM: DMA one 36KB Q tile (9216 f32, contiguous) global -> LDS --------
// D# layout per CDNA5 ISA ch.8: group0 = {flags, lds_addr, global_addr, type},
// group1 = {data_size, tensor_dim0, tile_dim0, stride}, groups 2/3 unused (1-D).
// This toolchain exposes the 6-arg builtin (extra int32x8 group slot + cpol).
__device__ __forceinline__ void tdm_load_q(const float* gsrc, float* ldst) {
  const unsigned long long ga = (unsigned long long)(uintptr_t)gsrc;
  const unsigned int la = (unsigned int)(uintptr_t)ldst;   // LDS byte offset
  u32x4 g0;
  g0[0] = 1u;                                              // count=1 (valid), user mode
  g0[1] = la;                                              // lds_addr [63:32]
  g0[2] = (unsigned int)(ga & 0xFFFFFFFFu);                // global_addr [95:64]
  g0[3] = (unsigned int)((ga >> 32) & 0x01FFFFFFu)         // global_addr [120:96]
        | (2u << 30);                                      // type=2 ("image") [127:126]
  i32x8 g1;
  g1[0] = (2 << 16);                // workgroup_mask=0 (no cluster), data_size=2 -> 4B
  g1[1] = (int)(9216u << 16);       // tensor_dim0 = 9216 elements (bits 63:48)
  g1[2] = (int)(1u << 16);          // tensor_dim0 hi=0, tensor_dim1 = 1
  g1[3] = (int)(9216u << 16);       // tile_dim0 = 9216 (bits 127:112)
  g1[4] = 0;                        // tile_dim1 = 0 (unused), tile_dim2 = 0
  g1[5] = 9216;                     // tensor_dim0_stride = 9216
  g1[6] = 0;                        // stride hi, tensor_dim1_stride lo
  g1[7] = 0;
  const i32x4 gz4 = {0, 0, 0, 0};   // groups 2/3 unused for 1-D/2-D tensors
  const i32x8 gz8 = {0, 0, 0, 0, 0, 0, 0, 0};
  __builtin_amdgcn_tensor_load_to_lds(g0, g1, gz4, gz4, gz8, 0);
}

// ---- WMMA helpers (V_WMMA_F32_16X16X4_F32, wave32) ------------------------
// A 16x4 f32 fragment: lane l -> row m = l&15, K pair base = (l>>4)*2
// B 4x16 / C,D 16x16: row-striped across lanes within a VGPR (ISA 7.12.2)
__device__ __forceinline__ v8f wmma_f32_16x16xK(
    const float* Abase, int lda, bool transA,
    const float* Bbase, int ldb, bool transB,
    int K, v8f acc, int lane)
{
  const int m  = lane & 15;          // row of A fragment / col of B fragment
  const int kq = (lane >> 4) << 1;   // 0 or 2
  for (int k = 0; k < K; k += 4) {
    const int k0 = k + kq;
    const int k1 = k + kq + 1;
    v2f a, b;
    if (!transA) { a = *(const v2fa*)(Abase + m * lda + k0); }   // ds_load_b64
    else         { a[0] = Abase[k0 * lda + m]; a[1] = Abase[k1 * lda + m]; }
    if (!transB) { b[0] = Bbase[k0 * ldb + m]; b[1] = Bbase[k1 * ldb + m]; }
    else         { b = *(const v2fa*)(Bbase + m * ldb + k0); }
    acc = __builtin_amdgcn_wmma_f32_16x16x4_f32(
        /*neg_a=*/false, a, /*neg_b=*/false, b,
        /*c_mod=*/(short)0, acc, /*reuse_a=*/false, /*reuse_b=*/false);
  }
  return acc;
}

__device__ __forceinline__ v8f load_tile16(const float* C, int ldc, int lane) {
  v8f c;
  const int n = lane & 15, r0 = (lane >> 4) * 8;
#pragma unroll
  for (int i = 0; i < 8; ++i) c[i] = C[(r0 + i) * ldc + n];
  return c;
}

__device__ __forceinline__ void store_tile16(float* D, int ldd, v8f d, int lane) {
  const int n = lane & 15, r0 = (lane >> 4) * 8;
#pragma unroll
  for (int i = 0; i < 8; ++i) D[(r0 + i) * ldd + n] = d[i];
}

// ---- Backward Riccati recursion: one workgroup (8 wave32) per batch -------
__global__ __launch_bounds__(256)
void lqr_backward(const float* __restrict__ Q, const float* __restrict__ p,
                  const float* __restrict__ A, const float* __restrict__ Bm,
                  const float* __restrict__ c1g,
                  float* __restrict__ Kg, float* __restrict__ kg)
{
  __shared__ float sF[NSd * NSCd];                 // F = [A | B], 64x96
  __shared__ float sV[NSd * NSd];                  // V, 64x64
  __shared__ float sv[NSd];
  __shared__ float sc1[NSd];
  __shared__ __align__(16) float sQbuf[2][NSCd * NSCd]; // TDM double buffer; Qh in place
  __shared__ float sp[NSCd];
  __shared__ float sFtV[NSCd * NSd];               // F^T V (96x64); reused as W (32x64)
  __shared__ float sqh[NSCd];
  __shared__ float sL[NCd * NCd];                  // Cholesky factor of Quu
  __shared__ float sK[NCd * NSd];                  // K (32x64)
  __shared__ float skk[NCd];
  __shared__ float stmp[NCd];

  const int b    = blockIdx.x;
  const int tid  = threadIdx.x;
  const int lane = tid & 31;
  const int wave = tid >> 5;

  // stage F = [A | B], c1; V = 0, v = 0
  for (int i = tid; i < NSd * NSd; i += 256) {
    const int s = i >> 6, j = i & 63;
    sF[s * NSCd + j] = A[(size_t)b * NSd * NSd + i];
  }
  for (int i = tid; i < NSd * NCd; i += 256) {
    const int s = i >> 5, j = i & 31;
    sF[s * NSCd + NSd + j] = Bm[(size_t)b * NSd * NCd + i];
  }
  for (int i = tid; i < NSd * NSd; i += 256) sV[i] = 0.f;
  if (tid < NSd) { sv[tid] = 0.f; sc1[tid] = c1g[b * NSd + tid]; }

  // prime the pipeline: DMA Q[T-1] into buffer (T-1)&1
  if (wave == 0)
    tdm_load_q(Q + (size_t)(b * TT + (TT - 1)) * (NSCd * NSCd), sQbuf[(TT - 1) & 1]);
  __syncthreads();

  for (int t = TT - 1; t >= 0; --t) {
    float* sQ = sQbuf[t & 1];
    // issue DMA for next step into the other buffer, then wait for current
    if (wave == 0) {
      if (t > 0) {
        tdm_load_q(Q + (size_t)(b * TT + (t - 1)) * (NSCd * NSCd), sQbuf[(t - 1) & 1]);
        __builtin_amdgcn_s_wait_tensorcnt(1);   // oldest (current buffer) landed
      } else {
        __builtin_amdgcn_s_wait_tensorcnt(0);
      }
    }
    if (tid < NSCd) sp[tid] = p[(b * TT + t) * NSCd + tid];
    __syncthreads();

    // ---- FtV[j][k] = sum_s F[s][j] V[s][k]  (96x64, K=64): 24 WMMA tiles ----
    for (int tile = wave; tile < 24; tile += 8) {
      const int jt = tile >> 2, kt = tile & 3;
      v8f acc = {};
      acc = wmma_f32_16x16xK(&sF[jt * 16], NSCd, true,
                             &sV[kt * 16], NSd, false, NSd, acc, lane);
      store_tile16(&sFtV[(jt * 16) * NSd + kt * 16], NSd, acc, lane);
    }
    __syncthreads();

    // ---- Qh = Q_t + FtV @ F  (96x96, K=64) in place: 36 WMMA tiles ----
    for (int tile = wave; tile < 36; tile += 8) {
      const int jt = tile / 6, it = tile % 6;
      v8f acc = load_tile16(&sQ[(jt * 16) * NSCd + it * 16], NSCd, lane);
      acc = wmma_f32_16x16xK(&sFtV[(jt * 16) * NSd], NSd, false,
                             &sF[it * 16], NSCd, false, NSd, acc, lane);
      store_tile16(&sQ[(jt * 16) * NSCd + it * 16], NSCd, acc, lane);
    }
    // qh[j] = p[j] + FtV[j][:].c1 + F[:][j].v
    if (tid < NSCd) {
      float acc = sp[tid];
      for (int k = 0; k < NSd; ++k) acc += sFtV[tid * NSd + k] * sc1[k];
      for (int s = 0; s < NSd; ++s) acc += sF[s * NSCd + tid] * sv[s];
      sqh[tid] = acc;
    }
    __syncthreads();

    // ---- Cholesky of Quu (32x32) entirely in wave 0 registers (wave32 shfl) ----
    if (wave == 0) {
      float row[NCd];                               // lane r holds row r of Quu
#pragma unroll
      for (int c = 0; c < NCd; ++c) row[c] = sQ[(NSd + lane) * NSCd + NSd + c];
#pragma unroll
      for (int k = 0; k < NCd; ++k) {
        const float dkk = __shfl(row[k], k, 32);
        const float dk  = sqrtf(dkk);
        const float lrk = row[k] / dk;              // lane r: L[r][k] (r>=k valid)
        row[k] = lrk;
#pragma unroll
        for (int c2 = k + 1; c2 < NCd; ++c2) {
          const float lck = __shfl(lrk, c2, 32);
          row[c2] -= lrk * lck;                     // trailing update
        }
      }
#pragma unroll
      for (int c = 0; c < NCd; ++c) sL[lane * NCd + c] = row[c];
    }
    __syncthreads();

    // ---- solve Quu X = [Qux | qu];  K = -X : threads 0..64, one RHS each ----
    if (tid < NSd + 1) {
      float y[NCd];
#pragma unroll
      for (int r = 0; r < NCd; ++r)
        y[r] = (tid < NSd) ? sQ[(NSd + r) * NSCd + tid] : sqh[NSd + r];
#pragma unroll
      for (int r = 0; r < NCd; ++r) {               // L y = rhs
        float s = y[r];
#pragma unroll
        for (int k2 = 0; k2 < NCd; ++k2)
          if (k2 < r) s -= sL[r * NCd + k2] * y[k2];
        y[r] = s / sL[r * NCd + r];
      }
#pragma unroll
      for (int r = NCd - 1; r >= 0; --r) {          // L^T x = y
        float s = y[r];
#pragma unroll
        for (int k2 = 0; k2 < NCd; ++k2)
          if (k2 > r) s -= sL[k2 * NCd + r] * y[k2];
        y[r] = s / sL[r * NCd + r];
      }
      if (tid < NSd) {
#pragma unroll
        for (int r = 0; r < NCd; ++r) sK[r * NSd + tid] = -y[r];
      } else {
#pragma unroll
        for (int r = 0; r < NCd; ++r) skk[r] = -y[r];
      }
    }
    __syncthreads();

    // spill K_t, k_t for the forward rollout
    {
      float* Ko = Kg + (size_t)(b * TT + t) * (NCd * NSd);
      for (int i = tid; i < NCd * NSd; i += 256) Ko[i] = sK[i];
      if (tid < NCd) kg[(b * TT + t) * NCd + tid] = skk[tid];
    }

    // ---- W = Quu @ K  (32x64, K=32) into reused sFtV space ----
    float* sW = sFtV;
    for (int tile = wave; tile < 8; tile += 8) {
      const int ut = tile >> 2, st = tile & 3;
      v8f acc = {};
      acc = wmma_f32_16x16xK(&sQ[(NSd + ut * 16) * NSCd + NSd], NSCd, false,
                             &sK[st * 16], NSd, false, NCd, acc, lane);
      store_tile16(&sW[(ut * 16) * NSd + st * 16], NSd, acc, lane);
    }
    // tmp[u] = qu[u] + Quu[u][:].kk
    if (tid < NCd) {
      float acc = sqh[NSd + tid];
      for (int w = 0; w < NCd; ++w) acc += sQ[(NSd + tid) * NSCd + NSd + w] * skk[w];
      stmp[tid] = acc;
    }
    __syncthreads();

    // ---- Vn = Qxx + Qxu K + K^T Qux + K^T W  (64x64): 16 tiles x 3 gemms ----
    for (int tile = wave; tile < 16; tile += 8) {
      const int mt = tile >> 2, nt = tile & 3;
      v8f acc = load_tile16(&sQ[(mt * 16) * NSCd + nt * 16], NSCd, lane);
      acc = wmma_f32_16x16xK(&sQ[(mt * 16) * NSCd + NSd], NSCd, false,
                             &sK[nt * 16], NSd, false, NCd, acc, lane);
      acc = wmma_f32_16x16xK(&sK[mt * 16], NSd, true,
                             &sQ[NSd * NSCd + nt * 16], NSCd, false, NCd, acc, lane);
      acc = wmma_f32_16x16xK(&sK[mt * 16], NSd, true,
                             &sW[nt * 16], NSd, false, NCd, acc, lane);
      store_tile16(&sV[(mt * 16) * NSd + nt * 16], NSd, acc, lane);
    }
    // vn[s] = qx[s] + Qxu[s][:].kk + sum_u K[u][s] * tmp[u]
    if (tid < NSd) {
      float acc = sqh[tid];
      for (int u = 0; u < NCd; ++u) acc += sQ[tid * NSCd + NSd + u] * skk[u];
      for (int u = 0; u < NCd; ++u) acc += sK[u * NSd + tid] * stmp[u];
      sv[tid] = acc;
    }
    __syncthreads();
  }
}

// ---- Forward rollout: one workgroup per batch -----------------------------
__global__ __launch_bounds__(256)
void lqr_forward(const float* __restrict__ Q, const float* __restrict__ p,
                 const float* __restrict__ A, const float* __restrict__ Bm,
                 const float* __restrict__ c1g, const float* __restrict__ x0,
                 const float* __restrict__ Kg, const float* __restrict__ kg,
                 float* __restrict__ xout, float* __restrict__ uout,
                 float* __restrict__ cost)
{
  __shared__ float sA[NSd * NSd];
  __shared__ float sB[NSd * NCd];
  __shared__ float sc1[NSd];
  __shared__ float sx[NSd];
  __shared__ float stau[NSCd];
  __shared__ float sKt[NCd * NSd];
  __shared__ float skt[NCd];
  __shared__ float sred[8];
  __shared__ float scost;

  const int b    = blockIdx.x;
  const int tid  = threadIdx.x;
  const int lane = tid & 31;
  const int wave = tid >> 5;

  for (int i = tid; i < NSd * NSd; i += 256) sA[i] = A[(size_t)b * NSd * NSd + i];
  for (int i = tid; i < NSd * NCd; i += 256) sB[i] = Bm[(size_t)b * NSd * NCd + i];
  if (tid < NSd) { sc1[tid] = c1g[b * NSd + tid]; sx[tid] = x0[b * NSd + tid]; }
  if (tid == 0) scost = 0.f;
  __syncthreads();

  for (int t = 0; t < TT; ++t) {
    const float* Ksrc = Kg + (size_t)(b * TT + t) * (NCd * NSd);
    for (int i = tid; i < NCd * NSd; i += 256) sKt[i] = Ksrc[i];
    if (tid < NCd) skt[tid] = kg[(b * TT + t) * NCd + tid];
    // prefetch next step's Q tile into cache (288 lines of 128B)
    if (t + 1 < TT) {
      const float* Qn = Q + (size_t)(b * TT + t + 1) * (NSCd * NSCd);
      __builtin_prefetch(Qn + tid * 32, 0, 1);
      if (tid < 32) __builtin_prefetch(Qn + (256 + tid) * 32, 0, 1);
    }
    __syncthreads();

    // u = K x + k ; tau = [x; u]
    if (tid < NCd) {
      float acc = skt[tid];
      for (int s = 0; s < NSd; ++s) acc += sKt[tid * NSd + s] * sx[s];
      stau[NSd + tid] = acc;
    }
    if (tid < NSd) stau[tid] = sx[tid];
    __syncthreads();

    if (tid < NSd) xout[(size_t)(b * TT + t) * NSd + tid] = stau[tid];
    if (tid < NCd) uout[(size_t)(b * TT + t) * NCd + tid] = stau[NSd + tid];

    // c = 0.5 tau^T Q tau + p.tau  (flat coalesced read of Q_t)
    const float* Qt = Q + (size_t)(b * TT + t) * (NSCd * NSCd);
    const float* pt = p + (b * TT + t) * NSCd;
    float part = 0.f;
    for (int e = tid; e < NSCd * NSCd; e += 256) {
      const int i = e / NSCd, j = e - i * NSCd;
      part += 0.5f * Qt[e] * stau[i] * stau[j];
    }
    if (tid < NSCd) part += pt[tid] * stau[tid];
    for (int off = 16; off > 0; off >>= 1) part += __shfl_down(part, off, 32);
    if (lane == 0) sred[wave] = part;
    __syncthreads();
    if (tid == 0) {
      float c = 0.f;
      for (int w = 0; w < 8; ++w) c += sred[w];
      scost += c;
    }

    // x_{t+1} = A x + B u + c1
    float xn = 0.f;
    if (tid < NSd) {
      xn = sc1[tid];
      for (int j = 0; j < NSd; ++j) xn += sA[tid * NSd + j] * stau[j];
      for (int j = 0; j < NCd; ++j) xn += sB[tid * NCd + j] * stau[NSd + j];
    }
    __syncthreads();
    if (tid < NSd) sx[tid] = xn;
    __syncthreads();
  }
  if (tid == 0) cost[b] = scost;
}

extern "C" void kernel_launch(void* const* d_in, const int* in_sizes, int n_in,
                              void* d_out, int out_size, void* d_ws, size_t ws_size,
                              hipStream_t stream) {
  (void)in_sizes; (void)n_in; (void)out_size; (void)ws_size;
  const float* Q      = (const float*)d_in[0];
  const float* p      = (const float*)d_in[1];
  const float* A      = (const float*)d_in[2];
  const float* B      = (const float*)d_in[3];
  const float* c1     = (const float*)d_in[4];
  const float* x_init = (const float*)d_in[5];

  float* xout = (float*)d_out;                               // (32,256,64)
  float* uout = xout + (size_t)NBATCH * TT * NSd;            // (32,256,32)
  float* cost = uout + (size_t)NBATCH * TT * NCd;            // (32,)

  float* Kg = (float*)d_ws;                                  // (32,256,32,64)
  float* kg = Kg + (size_t)NBATCH * TT * NCd * NSd;          // (32,256,32)

  lqr_backward<<<NBATCH, 256, 0, stream>>>(Q, p, A, B, c1, Kg, kg);
  lqr_forward<<<NBATCH, 256, 0, stream>>>(Q, p, A, B, c1, x_init, Kg, kg,
                                          xout, uout, cost);
}